// HamiltonianVersorNN_12713103196455
// MI455X (gfx1250) — compile-verified
//
#include <hip/hip_runtime.h>

// HamiltonianVersorNN on MI455X (gfx1250, wave32).
//
// Exact math reduction: the versor activation gates all blades by
// sigmoid(blade0) and the energy reads only blade0, so the gradient wrt x is
// supported on the blade-0 columns of every layer. fwd+bwd collapses to a
// 6->8->32->32->1 scalar MLP gradient per point (131072 points).
// fp32 WMMA (v_wmma_f32_16x16x4_f32) keeps it exact.
//
// Per wave: a 16-point tile; 6 GEMMs (embed, W1, W2, W2^T, W1^T, W_in^T) as
// 16x16x4 f32 WMMA chains (48 WMMAs/wave). All weights/biases are staged once
// per block into LDS, pre-transposed and zero-padded, so every A-fragment /
// bias access is an unconditional LDS load (no masking, no 64-bit address
// math in the wave body). Inter-GEMM relayout uses per-wave LDS tiles
// (row stride 17); odd weight strides (33/9) keep row-gathers conflict-free.

#define EMBED   8
#define HIDDEN  32
#define BLADES  32
#define DT      0.01f
#define TS      17            // activation tile row stride (floats)
#define WPB     8             // waves per block (256 threads)

// Shared-memory layout (float offsets). First four are read via ds_load_b128
// (offsets are multiples of 16 floats where needed for 16B alignment).
#define BIN0   0                      // 16  (bin blade-0, rows 8..15 zero)
#define B1C    16                     // 32  (b1[:,0])
#define B2C    48                     // 32  (b2[:,0])
#define W3C    80                     // 32  (W3[0,:])
#define W2S    112                    // 32x33  W2 row-major
#define W2T    (W2S + 32 * 33)        // 32x33  W2 transposed
#define W1S    (W2T + 32 * 33)        // 32x9   W1 row-major
#define W1T    (W1S + 32 * 9)         // 16x33  W1 transposed, rows 8..15 zero
#define WI0T   (W1T + 16 * 33)        // 16x9   Win0^T (i,d), pads zero
#define WI0S   (WI0T + 16 * 9)        // 16x9   Win0   (d,e), rows 6..15 zero
#define TILES  (WI0S + 16 * 9)        // per-wave tiles: WPB * 2 * 32 * TS
#define SMEMSZ (TILES + WPB * 2 * 32 * TS)

typedef __attribute__((ext_vector_type(2))) float v2f;
typedef __attribute__((ext_vector_type(8))) float v8f;

__device__ __forceinline__ v8f vzero8() {
  v8f z;
#pragma unroll
  for (int i = 0; i < 8; ++i) z[i] = 0.0f;
  return z;
}

__device__ __forceinline__ v8f wmma4(v2f a, v2f b, v8f c) {
  // D = A(16x4,f32) * B(4x16,f32) + C(16x16,f32)
  return __builtin_amdgcn_wmma_f32_16x16x4_f32(
      /*neg_a=*/false, a, /*neg_b=*/false, b,
      /*c_mod=*/(short)0, c, /*reuse_a=*/false, /*reuse_b=*/false);
}

__device__ __forceinline__ float sigm(float v) {
  return 1.0f / (1.0f + __expf(-v));
}

// A fragment (16x4 tile) from a staged LDS weight array, row stride ld.
// Lane layout (ISA 7.12.2): row = lane%16, cols = 2*(lane/16) + {0,1}.
// Arrays are pre-padded with zeros, so loads are unconditional.
__device__ __forceinline__ v2f load_afrag(const float* T, int ld, int row0,
                                          int k0, int lane) {
  const float* p = T + (row0 + (lane & 15)) * ld + k0 + 2 * (lane >> 4);
  v2f a;
  a[0] = p[0];
  a[1] = p[1];
  return a;
}

// B fragment (4x16 tile, rows k0..k0+3) from an activation tile (stride TS).
// Lane: col = lane%16, rows = k0 + 2*(lane/16) + {0,1}.
__device__ __forceinline__ v2f load_bfrag(const float* T, int k0, int lane) {
  int n = lane & 15;
  int r = k0 + 2 * (lane >> 4);
  v2f b;
  b[0] = T[r * TS + n];
  b[1] = T[(r + 1) * TS + n];
  return b;
}

// Store a 16x16 D fragment into an activation tile (stride TS).
__device__ __forceinline__ void store_dfrag(float* T, v8f d, int lane) {
  int n = lane & 15;
  int rb = 8 * (lane >> 4);
#pragma unroll
  for (int r = 0; r < 8; ++r) T[(rb + r) * TS + n] = d[r];
}

// Load 8 consecutive floats from LDS (two ds_load_b128; same-address within a
// half-wave -> LDS broadcast).
__device__ __forceinline__ void ld8(const float* p, float* v) {
  float4 a = ((const float4*)p)[0];
  float4 b = ((const float4*)p)[1];
  v[0] = a.x; v[1] = a.y; v[2] = a.z; v[3] = a.w;
  v[4] = b.x; v[5] = b.y; v[6] = b.z; v[7] = b.w;
}

__global__ __launch_bounds__(WPB * 32) void hvnn_kernel(
    const float* __restrict__ x,   const float* __restrict__ Win,
    const float* __restrict__ bin, const float* __restrict__ W1,
    const float* __restrict__ b1,  const float* __restrict__ W2,
    const float* __restrict__ b2,  const float* __restrict__ W3,
    float* __restrict__ out, int npoints) {
  __shared__ float sm[SMEMSZ];
  const int t = threadIdx.x;

  // ---- stage weights/biases once per block (pre-transposed, zero-padded) ---
  {
    // W2 (1024 floats): one float4 per thread -> W2S and W2T.
    float4 w = ((const float4*)W2)[t];
    int r = t >> 3, c0 = (t & 7) * 4;
#pragma unroll
    for (int j = 0; j < 4; ++j) {
      float wv = (&w.x)[j];
      sm[W2S + r * 33 + (c0 + j)] = wv;
      sm[W2T + (c0 + j) * 33 + r] = wv;
    }
    if (t < 64) {                     // W1 (256 floats) -> W1S and W1T
      float4 w1 = ((const float4*)W1)[t];
      int o = t >> 1, i0 = (t & 1) * 4;
#pragma unroll
      for (int j = 0; j < 4; ++j) {
        float wv = (&w1.x)[j];
        sm[W1S + o * 9 + (i0 + j)] = wv;
        sm[W1T + (i0 + j) * 33 + o] = wv;
      }
    }
    sm[W1T + 264 + t] = 0.0f;         // W1T rows 8..15 zero (264..519)
    if (t < 8) sm[W1T + 264 + 256 + t] = 0.0f;
    if (t < 128) {
      int i = t >> 3, d = t & 7;      // Win0^T (i,d): A of GEMM0
      sm[WI0T + i * 9 + d] =
          (i < EMBED && d < 6) ? Win[d * (EMBED * BLADES) + i * BLADES] : 0.0f;
      int dd = i, e = d;              // Win0 (d,e): A of GX
      sm[WI0S + dd * 9 + e] =
          (dd < 6) ? Win[dd * (EMBED * BLADES) + e * BLADES] : 0.0f;
    }
    if (t < 16) sm[BIN0 + t] = (t < EMBED) ? bin[t * BLADES] : 0.0f;
    if (t < 32) {
      sm[B1C + t] = b1[t * BLADES];
      sm[B2C + t] = b2[t * BLADES];
      sm[W3C + t] = W3[t];
    }
  }
  __syncthreads();

  const int lane = t & 31;
  const int wave = t >> 5;
  float* tA = &sm[TILES + wave * (2 * 32 * TS)];
  float* tB = tA + 32 * TS;

  const int tile = blockIdx.x * WPB + wave;
  const int P = tile * 16;
  if (P >= npoints) return;            // uniform per wave (after the barrier)

  const int n    = lane & 15;          // point column within tile
  const int half = lane >> 4;
  const int rb   = 8 * half;           // D-fragment row base

  // Out-of-range points: clamp the address; results are never stored.
  const int pt  = P + n;
  const int ptc = (pt < npoints) ? pt : (npoints - 1);
  const float* __restrict__ xp = x + (long)ptc * 6;  // 8B-aligned (24B stride)

  // ---- GEMM0: H0(8x16) = Win0^T(8x6) @ X(6x16) ----
  {
    v8f c = vzero8();
#pragma unroll
    for (int k0 = 0; k0 < 8; k0 += 4) {
      v2f a = load_afrag(sm + WI0T, 9, 0, k0, lane);
      // B[d][p] = x[pt*6+d]; d even -> aligned float2. d==6 is clamped to 4;
      // the A columns it multiplies are stored zero, so the value is ignored.
      int d  = k0 + 2 * half;
      int dc = (d < 6) ? d : 4;
      float2 xv = *(const float2*)(xp + dc);
      v2f b; b[0] = xv.x; b[1] = xv.y;
      c = wmma4(a, b, c);
    }
    float bb[8]; ld8(sm + BIN0 + rb, bb);   // rows 8..15 add zero
#pragma unroll
    for (int r = 0; r < 8; ++r) c[r] += bb[r];
    store_dfrag(tA, c, lane);
  }

  // ---- Layer 1: A1 = W1(32x8) @ H0 + b1[:,0]; z1, ds1 ----
  v8f ds1[2];
#pragma unroll
  for (int m = 0; m < 2; ++m) {
    v8f c = vzero8();
#pragma unroll
    for (int k0 = 0; k0 < EMBED; k0 += 4) {
      v2f a = load_afrag(sm + W1S, 9, m * 16, k0, lane);
      v2f b = load_bfrag(tA, k0, lane);
      c = wmma4(a, b, c);
    }
    float bb[8]; ld8(sm + B1C + m * 16 + rb, bb);
#pragma unroll
    for (int r = 0; r < 8; ++r) {
      float v = c[r] + bb[r];
      float s = sigm(v);
      c[r] = v * s;                                   // z1
      ds1[m][r] = s * (1.0f + v * (1.0f - s));        // dz1/da1
    }
    store_dfrag(tB + m * 16 * TS, c, lane);
  }

  // ---- Layer 2: A2 = W2(32x32) @ Z1 + b2[:,0]; g2 = W3 * dZ2/dA2 ----
#pragma unroll
  for (int m = 0; m < 2; ++m) {
    v8f c = vzero8();
#pragma unroll
    for (int k0 = 0; k0 < HIDDEN; k0 += 4) {
      v2f a = load_afrag(sm + W2S, 33, m * 16, k0, lane);
      v2f b = load_bfrag(tB, k0, lane);
      c = wmma4(a, b, c);
    }
    float bb[8], ww[8];
    ld8(sm + B2C + m * 16 + rb, bb);
    ld8(sm + W3C + m * 16 + rb, ww);
#pragma unroll
    for (int r = 0; r < 8; ++r) {
      float v = c[r] + bb[r];
      float s = sigm(v);
      c[r] = ww[r] * s * (1.0f + v * (1.0f - s));     // g2
    }
    store_dfrag(tA + m * 16 * TS, c, lane);
  }

  // ---- Backward: G1 = (W2^T @ G2) * ds1 ----
#pragma unroll
  for (int m = 0; m < 2; ++m) {
    v8f c = vzero8();
#pragma unroll
    for (int k0 = 0; k0 < HIDDEN; k0 += 4) {
      v2f a = load_afrag(sm + W2T, 33, m * 16, k0, lane);
      v2f b = load_bfrag(tA, k0, lane);
      c = wmma4(a, b, c);
    }
#pragma unroll
    for (int r = 0; r < 8; ++r) c[r] *= ds1[m][r];
    store_dfrag(tB + m * 16 * TS, c, lane);
  }

  // ---- Backward: GH0(8x16) = W1^T(8x32, rows 8..15 zero) @ G1 ----
  {
    v8f c = vzero8();
#pragma unroll
    for (int k0 = 0; k0 < HIDDEN; k0 += 4) {
      v2f a = load_afrag(sm + W1T, 33, 0, k0, lane);
      v2f b = load_bfrag(tB, k0, lane);
      c = wmma4(a, b, c);
    }
    store_dfrag(tA, c, lane);
  }

  // ---- Backward: GX(6x16) = Win0(6x8, rows 6..15 zero) @ GH0 ----
  v8f gx = vzero8();
#pragma unroll
  for (int k0 = 0; k0 < EMBED; k0 += 4) {
    v2f a = load_afrag(sm + WI0S, 9, 0, k0, lane);
    v2f b = load_bfrag(tA, k0, lane);
    gx = wmma4(a, b, gx);
  }

  // ---- Symplectic update: lanes 0..15 hold rows 0..7 of GX for point n ----
  if (half == 0 && pt < npoints) {
    const float2* xp2 = (const float2*)xp;
    float2* op2 = (float2*)(out + (long)pt * 6);
    float2 x01 = xp2[0], x23 = xp2[1], x45 = xp2[2];
    float2 o01 = make_float2(x01.x + gx[3] * DT, x01.y + gx[4] * DT);
    float2 o23 = make_float2(x23.x + gx[5] * DT, x23.y - gx[0] * DT);
    float2 o45 = make_float2(x45.x - gx[1] * DT, x45.y - gx[2] * DT);
    op2[0] = o01; op2[1] = o23; op2[2] = o45;
  }
}

extern "C" void kernel_launch(void* const* d_in, const int* in_sizes, int n_in,
                              void* d_out, int out_size, void* d_ws, size_t ws_size,
                              hipStream_t stream) {
  const float* x    = (const float*)d_in[0];
  const float* W_in = (const float*)d_in[1];
  const float* b_in = (const float*)d_in[2];
  const float* W1   = (const float*)d_in[3];
  const float* b1   = (const float*)d_in[4];
  const float* W2   = (const float*)d_in[5];
  const float* b2   = (const float*)d_in[6];
  const float* W3   = (const float*)d_in[7];
  // d_in[8] = b3: adds only a constant to the energy -> no gradient, unused.
  float* out = (float*)d_out;

  int npoints = in_sizes[0] / 6;                 // 131072
  int tiles   = (npoints + 15) / 16;             // 16 points per wave
  int blocks  = (tiles + WPB - 1) / WPB;         // 8 waves per block

  hvnn_kernel<<<blocks, WPB * 32, 0, stream>>>(x, W_in, b_in, W1, b1, W2, b2,
                                               W3, out, npoints);
}